// BBoxTransform_40407052320793
// MI455X (gfx1250) — compile-verified
//
#include <hip/hip_runtime.h>
#include <math.h>

typedef __attribute__((ext_vector_type(4))) float v4f;

#define TPB 256

// -------- Phase 1: per source box, precompute {x1, y1, x4, y4} --------------
__global__ __launch_bounds__(TPB) void bbox_phase1(
    const float* __restrict__ boxes, const float* __restrict__ deltas,
    v4f* __restrict__ ws, unsigned BN)
{
    unsigned t = blockIdx.x * TPB + threadIdx.x;
    if (t >= BN) return;
    size_t p = (size_t)t * 5u;
    float b0 = boxes[p + 0], b1 = boxes[p + 1];
    float b2 = boxes[p + 2], b3 = boxes[p + 3];
    float d0 = deltas[p + 0] * 0.1f, d1 = deltas[p + 1] * 0.1f;
    float d2 = deltas[p + 2] * 0.2f, d3 = deltas[p + 3] * 0.2f;
    float w = b2 - b0, h = b3 - b1;
    float pcx = b0 + 0.5f * w + d0 * w;
    float pcy = b1 + 0.5f * h + d1 * h;
    float pw = expf(d2) * w;
    float ph = expf(d3) * h;
    v4f v;
    v.x = pcx - 0.5f * pw;   // x1 (== x2)
    v.y = pcy - 0.5f * ph;   // y1 (== y3)
    v.z = pcx + 0.5f * pw;   // x4 (== x3)
    v.w = pcy + 0.5f * ph;   // y4 (== y2)
    ws[t] = v;
}

// Gather one element of the scrambled pred_boxes matrix.
// comp: 0,1->x1  2,3->x4  4,6->y1  5,7->y4  8..11->1.0
__device__ __forceinline__ float pred_at(const float* __restrict__ wsf,
                                         unsigned comp0, unsigned rem0,
                                         unsigned o, unsigned BN)
{
    unsigned rem = rem0 + o;                       // o < 12 << BN: at most one wrap
    unsigned comp = comp0 + (rem >= BN ? 1u : 0u);
    rem = (rem >= BN) ? rem - BN : rem;
    if (comp >= 8u) return 1.0f;
    unsigned off = (comp < 4u) ? (comp & 2u) : (1u + 2u * (comp & 1u));
    return wsf[(size_t)rem * 4u + off];
}

// -------- Phase 2: scrambled gather + rotation-about-center -----------------
__global__ __launch_bounds__(TPB) void bbox_phase2(
    const float* __restrict__ boxes, const float* __restrict__ deltas,
    const float* __restrict__ wsf, float* __restrict__ out, unsigned BN)
{
    unsigned t = blockIdx.x * TPB + threadIdx.x;
    if (t >= BN) return;
    size_t p = (size_t)t * 5u;
    float b0 = boxes[p + 0], b1 = boxes[p + 1];
    float b2 = boxes[p + 2], b3 = boxes[p + 3], b4 = boxes[p + 4];
    float d0 = deltas[p + 0] * 0.1f, d1 = deltas[p + 1] * 0.1f;
    float d4 = deltas[p + 4];
    float w = b2 - b0, h = b3 - b1;
    float pcx = b0 + 0.5f * w + d0 * w;
    float pcy = b1 + 0.5f * h + d1 * h;
    float pa = atanf(d4) + b4;
    float sa, ca;
    sincosf(pa, &sa, &ca);
    // M = T * R * Tinv : rotation about (pcx, pcy); rows 0..1 only.
    float m02 = pcx - ca * pcx + sa * pcy;
    float m12 = pcy - sa * pcx - ca * pcy;

    unsigned fbase = t * 12u;          // flat index into the (12B, N) concat array
    unsigned comp0 = fbase / BN;       // single integer divide per thread
    unsigned rem0  = fbase - comp0 * BN;

    float X[4], Y[4];
#pragma unroll
    for (unsigned j = 0; j < 4; ++j) {
        float px = pred_at(wsf, comp0, rem0, j,      BN);  // row i=0
        float py = pred_at(wsf, comp0, rem0, 4u + j, BN);  // row i=1
        float pz = pred_at(wsf, comp0, rem0, 8u + j, BN);  // row i=2 (NOT always 1!)
        X[j] = ca * px - sa * py + m02 * pz;
        Y[j] = sa * px + ca * py + m12 * pz;
    }
    // out[b,n,j,r] layout: t*8 + j*2 + r ; streamed once -> non-temporal b128 stores
    v4f o0 = { X[0], Y[0], X[1], Y[1] };
    v4f o1 = { X[2], Y[2], X[3], Y[3] };
    v4f* op = (v4f*)out + (size_t)t * 2u;
    __builtin_nontemporal_store(o0, op);
    __builtin_nontemporal_store(o1, op + 1);
}

// -------- Fallback: fused, recompute-at-gather (no workspace needed) --------
__device__ __forceinline__ float pred_raw(const float* __restrict__ boxes,
                                          const float* __restrict__ deltas,
                                          unsigned comp0, unsigned rem0,
                                          unsigned o, unsigned BN)
{
    unsigned rem = rem0 + o;
    unsigned comp = comp0 + (rem >= BN ? 1u : 0u);
    rem = (rem >= BN) ? rem - BN : rem;
    if (comp >= 8u) return 1.0f;
    size_t p = (size_t)rem * 5u;
    float b0 = boxes[p + 0], b1 = boxes[p + 1];
    float b2 = boxes[p + 2], b3 = boxes[p + 3];
    float w = b2 - b0, h = b3 - b1;
    if (comp < 4u) {
        float pcx = b0 + 0.5f * w + deltas[p + 0] * 0.1f * w;
        float pw  = expf(deltas[p + 2] * 0.2f) * w;
        return (comp & 2u) ? pcx + 0.5f * pw : pcx - 0.5f * pw;
    } else {
        float pcy = b1 + 0.5f * h + deltas[p + 1] * 0.1f * h;
        float ph  = expf(deltas[p + 3] * 0.2f) * h;
        return (comp == 5u || comp == 7u) ? pcy + 0.5f * ph : pcy - 0.5f * ph;
    }
}

__global__ __launch_bounds__(TPB) void bbox_fused(
    const float* __restrict__ boxes, const float* __restrict__ deltas,
    float* __restrict__ out, unsigned BN)
{
    unsigned t = blockIdx.x * TPB + threadIdx.x;
    if (t >= BN) return;
    size_t p = (size_t)t * 5u;
    float b0 = boxes[p + 0], b1 = boxes[p + 1];
    float b2 = boxes[p + 2], b3 = boxes[p + 3], b4 = boxes[p + 4];
    float d0 = deltas[p + 0] * 0.1f, d1 = deltas[p + 1] * 0.1f;
    float d4 = deltas[p + 4];
    float w = b2 - b0, h = b3 - b1;
    float pcx = b0 + 0.5f * w + d0 * w;
    float pcy = b1 + 0.5f * h + d1 * h;
    float pa = atanf(d4) + b4;
    float sa, ca;
    sincosf(pa, &sa, &ca);
    float m02 = pcx - ca * pcx + sa * pcy;
    float m12 = pcy - sa * pcx - ca * pcy;

    unsigned fbase = t * 12u;
    unsigned comp0 = fbase / BN;
    unsigned rem0  = fbase - comp0 * BN;

    float X[4], Y[4];
#pragma unroll
    for (unsigned j = 0; j < 4; ++j) {
        float px = pred_raw(boxes, deltas, comp0, rem0, j,      BN);
        float py = pred_raw(boxes, deltas, comp0, rem0, 4u + j, BN);
        float pz = pred_raw(boxes, deltas, comp0, rem0, 8u + j, BN);
        X[j] = ca * px - sa * py + m02 * pz;
        Y[j] = sa * px + ca * py + m12 * pz;
    }
    v4f o0 = { X[0], Y[0], X[1], Y[1] };
    v4f o1 = { X[2], Y[2], X[3], Y[3] };
    v4f* op = (v4f*)out + (size_t)t * 2u;
    __builtin_nontemporal_store(o0, op);
    __builtin_nontemporal_store(o1, op + 1);
}

extern "C" void kernel_launch(void* const* d_in, const int* in_sizes, int n_in,
                              void* d_out, int out_size, void* d_ws, size_t ws_size,
                              hipStream_t stream)
{
    const float* boxes  = (const float*)d_in[0];
    const float* deltas = (const float*)d_in[1];
    float* out = (float*)d_out;

    unsigned BN = (unsigned)(in_sizes[0] / 5);         // B*N source boxes
    unsigned blocks = (BN + TPB - 1) / TPB;
    size_t need = (size_t)BN * 4u * sizeof(float);     // {x1,y1,x4,y4} per box

    if (d_ws && ws_size >= need) {
        bbox_phase1<<<blocks, TPB, 0, stream>>>(boxes, deltas, (v4f*)d_ws, BN);
        bbox_phase2<<<blocks, TPB, 0, stream>>>(boxes, deltas,
                                                (const float*)d_ws, out, BN);
    } else {
        bbox_fused<<<blocks, TPB, 0, stream>>>(boxes, deltas, out, BN);
    }
}